// OptimizedAttentionLayer_3856880631845
// MI455X (gfx1250) — compile-verified
//
#include <hip/hip_runtime.h>
#include <math.h>

#define B_  8
#define TE_ 512
#define TD_ 128
#define H_  256

typedef float v2f __attribute__((ext_vector_type(2)));
typedef float v8f __attribute__((ext_vector_type(8)));

// CDNA5 hardware tanh (V_TANH_F32): single transcendental op instead of the
// ~20-op ocml software tanh. This is the dominant operation of the whole
// problem (B*TD*TE*H = 134M evaluations), so it must be the native trans op.
__device__ __forceinline__ float fast_tanh(float x) {
#if defined(__has_builtin)
#if __has_builtin(__builtin_amdgcn_tanhf)
    return __builtin_amdgcn_tanhf(x);
#else
    float r;
    asm volatile("v_tanh_f32 %0, %1" : "=v"(r) : "v"(x));
    return r;
#endif
#else
    float r;
    asm volatile("v_tanh_f32 %0, %1" : "=v"(r) : "v"(x));
    return r;
#endif
}

// ---------------------------------------------------------------------------
// Generic row-major fp32 GEMM: C[M,N] = A[M,K] * W[K,N], one 16x16 tile/wave,
// inner product via V_WMMA_F32_16X16X4_F32. Optional batching via blockIdx.y.
// M, N, K must be multiples of 16 / 16 / 4 (true for all uses here).
// ---------------------------------------------------------------------------
__global__ void gemm_f32_wmma_kernel(const float* __restrict__ A,
                                     const float* __restrict__ W,
                                     float* __restrict__ C,
                                     int M, int N, int K,
                                     long long sA, long long sB, long long sC)
{
    const float* Ab = A + (long long)blockIdx.y * sA;
    const float* Wb = W + (long long)blockIdx.y * sB;
    float*       Cb = C + (long long)blockIdx.y * sC;

    const int lane   = threadIdx.x & 31;
    const int waveId = threadIdx.x >> 5;
    const int wavesPerBlock = blockDim.x >> 5;
    const int tile   = blockIdx.x * wavesPerBlock + waveId;
    const int tilesN = N >> 4;
    const int tm = (tile / tilesN) << 4;
    const int tn = (tile % tilesN) << 4;
    if (tm >= M) return;                 // wave-uniform: EXEC stays all-ones

    const int half = lane >> 4;          // 0: K pair {0,1}, 1: K pair {2,3}
    const int l16  = lane & 15;
    const long long arow = (long long)(tm + l16) * K;   // A row of this lane
    const int n = tn + l16;                             // W/C column of this lane

    v8f acc = {};
#pragma unroll 4
    for (int k0 = 0; k0 < K; k0 += 4) {
        const int ka = k0 + (half << 1);
        v2f a, b;
        // A 16x4 fragment: lane row m, elements K=ka, ka+1 (contiguous -> b64)
        a.x = Ab[arow + ka];
        a.y = Ab[arow + ka + 1];
        // B 4x16 fragment: lane col n, rows K=ka, ka+1
        b.x = Wb[(long long)ka * N + n];
        b.y = Wb[(long long)(ka + 1) * N + n];
        acc = __builtin_amdgcn_wmma_f32_16x16x4_f32(
                  /*neg_a=*/false, a, /*neg_b=*/false, b,
                  /*c_mod=*/(short)0, acc, /*reuse_a=*/false, /*reuse_b=*/false);
    }

    // C/D layout: VGPR v -> row tm + v + 8*half, col n
#pragma unroll
    for (int v = 0; v < 8; ++v)
        Cb[(long long)(tm + (half << 3) + v) * N + n] = acc[v];
}

// ---------------------------------------------------------------------------
// score[b,d,e] = sum_h V[h] * tanh(encp[b,e,h] + decp[b,d,h]); softmax over e.
// One 256-thread block per (b,d). Writes normalized weights to attn[b,d,:].
// LDS reads in the hot loop are same-address broadcasts (conflict-free).
// ---------------------------------------------------------------------------
__global__ void score_softmax_kernel(const float* __restrict__ encp,
                                     const float* __restrict__ decp,
                                     const float* __restrict__ Va,
                                     float* __restrict__ attn)
{
    const int bd = blockIdx.x;           // 0 .. B*TD-1
    const int b  = bd / TD_;

    __shared__ float s_dec[H_];
    __shared__ float s_v[H_];
    __shared__ float s_score[TE_];
    __shared__ float s_red[8];

    for (int h = threadIdx.x; h < H_; h += blockDim.x) {
        s_dec[h] = decp[(long long)bd * H_ + h];
        s_v[h]   = Va[h];
    }
    __syncthreads();

    const float* ep = encp + (long long)b * TE_ * H_;
    for (int e = threadIdx.x; e < TE_; e += blockDim.x) {
        const float* row = ep + (long long)e * H_;
        float s = 0.f;
#pragma unroll 8
        for (int h = 0; h < H_; ++h)
            s = fmaf(s_v[h], fast_tanh(row[h] + s_dec[h]), s);
        s_score[e] = s;
    }
    __syncthreads();

    const int lane = threadIdx.x & 31;
    const int wid  = threadIdx.x >> 5;
    const int nw   = blockDim.x >> 5;

    // --- block max ---
    float m = -INFINITY;
    for (int e = threadIdx.x; e < TE_; e += blockDim.x) m = fmaxf(m, s_score[e]);
    for (int off = 16; off > 0; off >>= 1) m = fmaxf(m, __shfl_down(m, off, 32));
    if (lane == 0) s_red[wid] = m;
    __syncthreads();
    if (threadIdx.x == 0) {
        float mm = s_red[0];
        for (int i = 1; i < nw; ++i) mm = fmaxf(mm, s_red[i]);
        s_red[0] = mm;
    }
    __syncthreads();
    m = s_red[0];
    __syncthreads();

    // --- exp + block sum ---
    float partial = 0.f;
    for (int e = threadIdx.x; e < TE_; e += blockDim.x) {
        float w = __expf(s_score[e] - m);
        s_score[e] = w;
        partial += w;
    }
    for (int off = 16; off > 0; off >>= 1) partial += __shfl_down(partial, off, 32);
    if (lane == 0) s_red[wid] = partial;
    __syncthreads();
    if (threadIdx.x == 0) {
        float ss = 0.f;
        for (int i = 0; i < nw; ++i) ss += s_red[i];
        s_red[0] = ss;
    }
    __syncthreads();
    const float inv = 1.0f / s_red[0];

    float* arow = attn + (long long)bd * TE_;
    for (int e = threadIdx.x; e < TE_; e += blockDim.x)
        arow[e] = s_score[e] * inv;
}

// ---------------------------------------------------------------------------
extern "C" void kernel_launch(void* const* d_in, const int* in_sizes, int n_in,
                              void* d_out, int out_size, void* d_ws, size_t ws_size,
                              hipStream_t stream)
{
    const float* enc = (const float*)d_in[0];   // [B,TE,H]
    const float* dec = (const float*)d_in[1];   // [B,TD,H]
    const float* Wa  = (const float*)d_in[2];   // [H,H]
    const float* Ua  = (const float*)d_in[3];   // [H,H]
    const float* Va  = (const float*)d_in[4];   // [H,1]

    float* ctx  = (float*)d_out;                       // [B,TD,H]  = 262144
    float* attn = ctx + (size_t)B_ * TD_ * H_;         // [B,TD,TE] = 524288

    float* encp = (float*)d_ws;                        // [B*TE, H]  4 MB
    float* decp = encp + (size_t)B_ * TE_ * H_;        // [B*TD, H]  1 MB

    // 1) enc_proj = enc @ W_a   (batch folded into M; W shared across batch)
    {
        const int M = B_ * TE_, N = H_, K = H_;
        const int tiles = (M / 16) * (N / 16);         // 4096
        gemm_f32_wmma_kernel<<<dim3(tiles / 4), dim3(128), 0, stream>>>(
            enc, Wa, encp, M, N, K, 0, 0, 0);
    }
    // 2) dec_proj = dec @ U_a
    {
        const int M = B_ * TD_, N = H_, K = H_;
        const int tiles = (M / 16) * (N / 16);         // 1024
        gemm_f32_wmma_kernel<<<dim3(tiles / 4), dim3(128), 0, stream>>>(
            dec, Ua, decp, M, N, K, 0, 0, 0);
    }
    // 3) score + softmax -> attention weights (second output slice)
    score_softmax_kernel<<<dim3(B_ * TD_), dim3(256), 0, stream>>>(
        encp, decp, Va, attn);
    // 4) context[b] = attn[b] @ enc[b]  (batched over grid.y)
    {
        const int M = TD_, N = H_, K = TE_;
        const int tiles = (M / 16) * (N / 16);         // 128
        gemm_f32_wmma_kernel<<<dim3(tiles / 4, B_), dim3(128), 0, stream>>>(
            attn, enc, ctx, M, N, K,
            (long long)TD_ * TE_, (long long)TE_ * H_, (long long)TD_ * H_);
    }
}